// VectorQuantizer_11854109737195
// MI455X (gfx1250) — compile-verified
//
#include <hip/hip_runtime.h>

typedef __attribute__((ext_vector_type(2))) float v2f;
typedef __attribute__((ext_vector_type(8))) float v8f;
typedef __attribute__((ext_vector_type(4))) float fvec4;

#define EMB_D 64
#define EMB_K 1024

// ---------------------------------------------------------------------------
// Kernel 1: ||e_k||^2 per codebook row (K floats into workspace)
// ---------------------------------------------------------------------------
__global__ __launch_bounds__(256) void vq_enorm_kernel(
    const float* __restrict__ emb, float* __restrict__ enorm, int K) {
  int k = blockIdx.x * blockDim.x + threadIdx.x;
  if (k >= K) return;
  const fvec4* e = (const fvec4*)(emb + (size_t)k * EMB_D);
  float s = 0.f;
#pragma unroll
  for (int i = 0; i < EMB_D / 4; ++i) {
    fvec4 v = e[i];
    s += v.x * v.x + v.y * v.y + v.z * v.z + v.w * v.w;
  }
  enorm[k] = s;
}

// ---------------------------------------------------------------------------
// Kernel 2: WMMA distance + argmin. One wave32 per 16-row tile.
// dist(n,k) = ||e_k||^2 - 2 * <x_n, e_k>   (||x_n||^2 dropped: row-constant)
// A-frag (16x4 f32): lane l -> row (l&15), dims 4*kk + 2*(l>=16) + {0,1}
// B-frag (4x16 f32): identical pattern with "row" = code index.
// C/D (16x16 f32):   VGPR r, lane l -> (M = r + 8*(l>=16), N = l&15)
// ---------------------------------------------------------------------------
__global__ __launch_bounds__(128) void vq_argmin_kernel(
    const float* __restrict__ x, const float* __restrict__ emb,
    const float* __restrict__ enorm, int* __restrict__ idxbuf, int nrows) {
  const int lane = threadIdx.x & 31;
  const int wave = (blockIdx.x * blockDim.x + threadIdx.x) >> 5;
  const int rowbase = wave * 16;
  if (rowbase >= nrows) return;  // wave-uniform

  const int half = lane >> 4;  // 0: dims {4k,4k+1}, 1: dims {4k+2,4k+3}
  const int lid = lane & 15;

  // Load the whole 16x64 A tile into registers: 16 float2 per lane.
  v2f a[16];
  const float* xr = x + (size_t)(rowbase + lid) * EMB_D + 2 * half;
#pragma unroll
  for (int kk = 0; kk < 16; ++kk) a[kk] = *(const v2f*)(xr + 4 * kk);

  float minv[8];
  int mini[8];
#pragma unroll
  for (int r = 0; r < 8; ++r) {
    minv[r] = 3.402823466e38f;
    mini[r] = 0;
  }

  for (int c0 = 0; c0 < EMB_K; c0 += 16) {
    const float* er = emb + (size_t)(c0 + lid) * EMB_D + 2 * half;
    v8f acc = {};
#pragma unroll
    for (int kk = 0; kk < 16; ++kk) {
      v2f b = *(const v2f*)(er + 4 * kk);
      // D = A(16x4) x B(4x16) + C ; accumulate x . e^T over D=64
      acc = __builtin_amdgcn_wmma_f32_16x16x4_f32(
          /*neg_a=*/false, a[kk], /*neg_b=*/false, b,
          /*c_mod=*/(short)0, acc, /*reuse_a=*/false, /*reuse_b=*/false);
    }
    const float en = enorm[c0 + lid];
    const int cand = c0 + lid;  // this lane's code column
#pragma unroll
    for (int r = 0; r < 8; ++r) {
      float dist = en - 2.0f * acc[r];
      if (dist < minv[r]) {  // strict <: keeps earliest index per lane
        minv[r] = dist;
        mini[r] = cand;
      }
    }
  }

  // Butterfly min-reduce across the 16 column-lanes (stays within each half).
#pragma unroll
  for (int off = 1; off < 16; off <<= 1) {
#pragma unroll
    for (int r = 0; r < 8; ++r) {
      float ov = __shfl_xor(minv[r], off, 32);
      int oi = __shfl_xor(mini[r], off, 32);
      if (ov < minv[r] || (ov == minv[r] && oi < mini[r])) {
        minv[r] = ov;
        mini[r] = oi;
      }
    }
  }

  if (lid == 0) {  // lane 0 -> rows 0..7, lane 16 -> rows 8..15
#pragma unroll
    for (int r = 0; r < 8; ++r) idxbuf[rowbase + half * 8 + r] = mini[r];
  }
}

// ---------------------------------------------------------------------------
// Kernel 3: gather quantized = embedding[idx] (== quantized_st numerically),
// emit per-block partial sums of (q - x)^2 for a deterministic reduction.
// One float4 per thread.
// ---------------------------------------------------------------------------
__global__ __launch_bounds__(256) void vq_gather_kernel(
    const float* __restrict__ x, const float* __restrict__ emb,
    const int* __restrict__ idxbuf, float* __restrict__ qout,
    float* __restrict__ partial, int nvec4) {
  __shared__ float red[256];
  int i = blockIdx.x * blockDim.x + threadIdx.x;
  float s = 0.f;
  if (i < nvec4) {
    int row = i >> 4;  // D/4 = 16 float4 per row
    int sub = i & 15;
    int idx = idxbuf[row];
    fvec4 q = ((const fvec4*)(emb + (size_t)idx * EMB_D))[sub];
    fvec4 xi = ((const fvec4*)x)[i];
    fvec4 d = q - xi;
    s = d.x * d.x + d.y * d.y + d.z * d.z + d.w * d.w;
    __builtin_nontemporal_store(q, (fvec4*)qout + i);
  }
  red[threadIdx.x] = s;
  __syncthreads();
  for (int o = 128; o > 0; o >>= 1) {
    if ((int)threadIdx.x < o) red[threadIdx.x] += red[threadIdx.x + o];
    __syncthreads();
  }
  if (threadIdx.x == 0) partial[blockIdx.x] = red[0];
}

// ---------------------------------------------------------------------------
// Kernel 4: one-hot encodings, 268 MB streamed with non-temporal b128 stores.
// One float4 per thread; codebook index per row re-read (hits L2).
// ---------------------------------------------------------------------------
__global__ __launch_bounds__(256) void vq_onehot_kernel(
    const int* __restrict__ idxbuf, float* __restrict__ enc, int nvec4) {
  int i = blockIdx.x * blockDim.x + threadIdx.x;
  if (i >= nvec4) return;
  int row = i >> 8;             // K/4 = 256 float4 per row
  int c0 = (i & 255) << 2;      // first code column of this float4
  int idx = idxbuf[row];
  fvec4 v;
  v.x = (idx == c0 + 0) ? 1.0f : 0.0f;
  v.y = (idx == c0 + 1) ? 1.0f : 0.0f;
  v.z = (idx == c0 + 2) ? 1.0f : 0.0f;
  v.w = (idx == c0 + 3) ? 1.0f : 0.0f;
  __builtin_nontemporal_store(v, (fvec4*)enc + i);
}

// ---------------------------------------------------------------------------
// Kernel 5: reduce partials -> mse; loss = 1.25*mse; codebook = commit = mse.
// ---------------------------------------------------------------------------
__global__ __launch_bounds__(256) void vq_finalize_kernel(
    const float* __restrict__ partial, int nparts, float* __restrict__ out3,
    float inv_count) {
  __shared__ float red[256];
  float s = 0.f;
  for (int i = threadIdx.x; i < nparts; i += 256) s += partial[i];
  red[threadIdx.x] = s;
  __syncthreads();
  for (int o = 128; o > 0; o >>= 1) {
    if ((int)threadIdx.x < o) red[threadIdx.x] += red[threadIdx.x + o];
    __syncthreads();
  }
  if (threadIdx.x == 0) {
    float mse = red[0] * inv_count;
    out3[0] = mse * 1.25f;  // loss = codebook + 0.25 * commitment
    out3[1] = mse;          // codebook_loss
    out3[2] = mse;          // commitment_loss
  }
}

// ---------------------------------------------------------------------------
extern "C" void kernel_launch(void* const* d_in, const int* in_sizes, int n_in,
                              void* d_out, int out_size, void* d_ws,
                              size_t ws_size, hipStream_t stream) {
  const float* x = (const float*)d_in[0];    // [16,4096,64]
  const float* emb = (const float*)d_in[1];  // [1024,64]
  const int N = in_sizes[0] / EMB_D;         // 65536
  const int K = in_sizes[1] / EMB_D;         // 1024

  // workspace layout
  float* enorm = (float*)d_ws;                                  // K floats
  int* idxbuf = (int*)((char*)d_ws + (size_t)K * sizeof(float));  // N ints
  float* partial =
      (float*)((char*)idxbuf + (size_t)N * sizeof(int));        // partial sums

  // output layout (flat concat in return order)
  float* qout = (float*)d_out;                 // [N, 64]
  float* enc = qout + (size_t)N * EMB_D;       // [N, K]
  float* out3 = enc + (size_t)N * K;           // 3 scalars

  // 1) codebook norms
  vq_enorm_kernel<<<(K + 255) / 256, 256, 0, stream>>>(emb, enorm, K);

  // 2) WMMA distances + argmin: N/16 waves, 4 waves per block
  {
    int waves = N / 16;
    int blocks = (waves * 32 + 127) / 128;
    vq_argmin_kernel<<<blocks, 128, 0, stream>>>(x, emb, enorm, idxbuf, N);
  }

  // 3) gather + loss partials
  int gvec4 = N * (EMB_D / 4);
  int gblocks = (gvec4 + 255) / 256;
  vq_gather_kernel<<<gblocks, 256, 0, stream>>>(x, emb, idxbuf, qout, partial,
                                                gvec4);

  // 4) one-hot encodings (NT streaming stores)
  int evec4 = N * (K / 4);
  vq_onehot_kernel<<<(evec4 + 255) / 256, 256, 0, stream>>>(idxbuf, enc, evec4);

  // 5) scalars
  vq_finalize_kernel<<<1, 256, 0, stream>>>(partial, gblocks, out3,
                                            1.0f / ((float)N * EMB_D));
}